// DGCNN_14061722927242
// MI455X (gfx1250) — compile-verified
//
#include <hip/hip_runtime.h>

// DGCNN forward for MI455X (gfx1250, wave32).
// Precision: pure fp32 (V_WMMA_F32_16X16X4_F32) — GEMMs are ~1.5 GFLOP total,
// the workload is bound by L2-resident gather/scatter (node features are
// 12.8 MB << 192 MB L2), so fp32 matrix ops are free and numerically exact.

typedef __attribute__((ext_vector_type(2))) float v2f;
typedef __attribute__((ext_vector_type(8))) float v8f;

#define HDIM 64

// ---------------- degree / norm ----------------
__global__ void deg_init_kernel(float* __restrict__ deg, int n) {
  int i = blockIdx.x * blockDim.x + threadIdx.x;
  if (i < n) deg[i] = 1.0f;  // self-loop contribution
}

__global__ void deg_count_kernel(const int* __restrict__ dst, float* __restrict__ deg, int e) {
  int i = blockIdx.x * blockDim.x + threadIdx.x;
  if (i < e) unsafeAtomicAdd(&deg[dst[i]], 1.0f);  // exact: small integer counts
}

__global__ void deg_rsqrt_kernel(float* __restrict__ deg, int n) {
  int i = blockIdx.x * blockDim.x + threadIdx.x;
  if (i < n) deg[i] = rsqrtf(fmaxf(deg[i], 1.0f));  // in-place deg -> dinv
}

// ---------------- GEMM: T = A[N,64] @ W[64,64], fp32 WMMA ----------------
// One wave computes a 16x64 tile: 4 N-tiles x 16 K-steps of v_wmma_f32_16x16x4_f32.
__global__ __launch_bounds__(128) void gcn_gemm64_kernel(
    const float* __restrict__ A, const float* __restrict__ W,
    float* __restrict__ T, int nrows) {
  __shared__ float Ws[HDIM * HDIM];  // 16 KB of the 320 KB LDS
  for (int i = threadIdx.x; i < HDIM * HDIM; i += 128) Ws[i] = W[i];
  __syncthreads();

  const int wave = threadIdx.x >> 5;           // 0..3
  const int lane = threadIdx.x & 31;
  const int tile = blockIdx.x * 4 + wave;      // 16-row tile index
  const int row0 = tile * 16;
  if (row0 >= nrows) return;
  const bool fullTile = (row0 + 16 <= nrows);  // wave-uniform

  // A fragment layout (16x4 f32): lane L holds row M=L&15, K pair k0=(L>>4)*2.
  const int m  = lane & 15;
  const int k0 = (lane >> 4) * 2;
  const int arow = (row0 + m < nrows) ? (row0 + m) : (nrows - 1);  // tail clamp
  const float* __restrict__ ap = A + (size_t)arow * HDIM;

  v2f afrag[16];
#pragma unroll
  for (int kk = 0; kk < 16; ++kk)
    afrag[kk] = *(const v2f*)(ap + kk * 4 + k0);  // 8B aligned (k0 even)

  const int n = lane & 15;
  v8f acc[4];
#pragma unroll
  for (int nt = 0; nt < 4; ++nt) {
    acc[nt] = (v8f){};
#pragma unroll
    for (int kk = 0; kk < 16; ++kk) {
      // B fragment (4x16 f32): lane L holds col N=L&15, K pair k0.
      v2f bfrag;
      bfrag.x = Ws[(kk * 4 + k0) * HDIM + nt * 16 + n];       // lowers to
      bfrag.y = Ws[(kk * 4 + k0 + 1) * HDIM + nt * 16 + n];   // ds_load_2addr_stride64
      acc[nt] = __builtin_amdgcn_wmma_f32_16x16x4_f32(
          /*neg_a=*/false, afrag[kk], /*neg_b=*/false, bfrag,
          /*c_mod=*/(short)0, acc[nt], /*reuse_a=*/false, /*reuse_b=*/false);
    }
  }

  // C/D layout (16x16 f32, 8 VGPRs): vgpr r -> M = (lane>>4)*8 + r, N = lane&15.
  const int mbase = (lane >> 4) * 8;
  if (fullTile) {
    // Fast path: straight-line stores, no per-element EXEC predication.
#pragma unroll
    for (int nt = 0; nt < 4; ++nt)
#pragma unroll
      for (int r = 0; r < 8; ++r)
        T[(size_t)(row0 + mbase + r) * HDIM + nt * 16 + n] = acc[nt][r];
  } else {
#pragma unroll
    for (int nt = 0; nt < 4; ++nt)
#pragma unroll
      for (int r = 0; r < 8; ++r)
        if (row0 + mbase + r < nrows)
          T[(size_t)(row0 + mbase + r) * HDIM + nt * 16 + n] = acc[nt][r];
  }
}

// ---------------- out = bias + self-loop message ----------------
__global__ void gcn_init64_kernel(const float* __restrict__ T,
                                  const float* __restrict__ dinv,
                                  const float* __restrict__ bias,
                                  float* __restrict__ O, int n) {
  int idx = blockIdx.x * blockDim.x + threadIdx.x;
  if (idx >= n * HDIM) return;
  const int i = idx >> 6;
  const int j = idx & 63;
  const float di = dinv[i];
  O[idx] = bias[j] + di * di * T[idx];
}

// ---------------- edge scatter: O[dst] += norm * T[src], 16 thr/edge ----------------
__global__ void gcn_scatter64_kernel(const int* __restrict__ src,
                                     const int* __restrict__ dst,
                                     const float* __restrict__ dinv,
                                     const float* __restrict__ T,
                                     float* __restrict__ O, int e) {
  int idx = blockIdx.x * blockDim.x + threadIdx.x;
  const int ei = idx >> 4;
  if (ei >= e) return;
  const int part = (idx & 15) * 4;
  const int s = src[ei];
  const int d = dst[ei];
  const float w = dinv[s] * dinv[d];
  const float4 v = *(const float4*)(T + (size_t)s * HDIM + part);  // 16B aligned
  float* o = O + (size_t)d * HDIM + part;
  unsafeAtomicAdd(o + 0, v.x * w);   // global_atomic_add_f32, L2-resident
  unsafeAtomicAdd(o + 1, v.y * w);
  unsafeAtomicAdd(o + 2, v.z * w);
  unsafeAtomicAdd(o + 3, v.w * w);
}

// ---------------- layer 4 (64 -> 1): GEMV + self-loop init ----------------
__global__ void layer4_kernel(const float* __restrict__ h3,
                              const float* __restrict__ W3,
                              const float* __restrict__ b3,
                              const float* __restrict__ dinv,
                              float* __restrict__ t1, float* __restrict__ h4, int n) {
  int i = blockIdx.x * blockDim.x + threadIdx.x;
  if (i >= n) return;
  const float* __restrict__ row = h3 + (size_t)i * HDIM;
  float acc = 0.0f;
#pragma unroll
  for (int j = 0; j < HDIM; ++j) acc = fmaf(row[j], W3[j], acc);
  t1[i] = acc;
  const float di = dinv[i];
  h4[i] = b3[0] + di * di * acc;
}

__global__ void scatter1_kernel(const int* __restrict__ src,
                                const int* __restrict__ dst,
                                const float* __restrict__ dinv,
                                const float* __restrict__ t1,
                                float* __restrict__ h4, int e) {
  int i = blockIdx.x * blockDim.x + threadIdx.x;
  if (i >= e) return;
  const int s = src[i];
  const int d = dst[i];
  unsafeAtomicAdd(&h4[d], dinv[s] * dinv[d] * t1[s]);
}

// ---------------- final Conv1d == 193 -> 16 projection ----------------
__global__ __launch_bounds__(256) void conv_kernel(
    const float* __restrict__ h1, const float* __restrict__ h2,
    const float* __restrict__ h3, const float* __restrict__ h4,
    const float* __restrict__ cw, const float* __restrict__ cb,
    float* __restrict__ out, int n) {
  __shared__ float sw[16 * 193];
  __shared__ float sb[16];
  for (int i = threadIdx.x; i < 16 * 193; i += 256) sw[i] = cw[i];
  if (threadIdx.x < 16) sb[threadIdx.x] = cb[threadIdx.x];
  __syncthreads();

  const int idx = blockIdx.x * 256 + threadIdx.x;
  const int i = idx >> 4;
  const int c = idx & 15;
  if (i >= n) return;
  const float* __restrict__ w = sw + c * 193;
  const float* __restrict__ r1 = h1 + (size_t)i * HDIM;
  const float* __restrict__ r2 = h2 + (size_t)i * HDIM;
  const float* __restrict__ r3 = h3 + (size_t)i * HDIM;
  float acc = sb[c];
#pragma unroll 8
  for (int j = 0; j < HDIM; ++j) acc = fmaf(r1[j], w[j], acc);
#pragma unroll 8
  for (int j = 0; j < HDIM; ++j) acc = fmaf(r2[j], w[64 + j], acc);
#pragma unroll 8
  for (int j = 0; j < HDIM; ++j) acc = fmaf(r3[j], w[128 + j], acc);
  acc = fmaf(h4[i], w[192], acc);
  out[idx] = acc;
}

// ---------------- launch ----------------
extern "C" void kernel_launch(void* const* d_in, const int* in_sizes, int n_in,
                              void* d_out, int out_size, void* d_ws, size_t ws_size,
                              hipStream_t stream) {
  const float* x  = (const float*)d_in[0];
  const int*   ei = (const int*)d_in[1];
  // d_in[2] = edge_attr (unused, use_edge_attr=False)
  const float* W0 = (const float*)d_in[3];
  const float* b0 = (const float*)d_in[4];
  const float* W1 = (const float*)d_in[5];
  const float* b1 = (const float*)d_in[6];
  const float* W2 = (const float*)d_in[7];
  const float* b2 = (const float*)d_in[8];
  const float* W3 = (const float*)d_in[9];
  const float* b3 = (const float*)d_in[10];
  const float* cw = (const float*)d_in[11];
  const float* cb = (const float*)d_in[12];
  float* out = (float*)d_out;

  const int N = in_sizes[0] / HDIM;
  const int E = in_sizes[1] / 2;
  const int* srcp = ei;      // edge_index[0]
  const int* dstp = ei + E;  // edge_index[1]

  // Workspace carve-up (fp32): dinv[N] tmp[N*64] h1[N*64] h2[N*64] h3[N*64] t1[N] h4[N]
  float* ws   = (float*)d_ws;
  float* dinv = ws;                 ws += N;
  float* tmp  = ws;                 ws += (size_t)N * HDIM;
  float* h1   = ws;                 ws += (size_t)N * HDIM;
  float* h2   = ws;                 ws += (size_t)N * HDIM;
  float* h3   = ws;                 ws += (size_t)N * HDIM;
  float* t1   = ws;                 ws += N;
  float* h4   = ws;                 ws += N;

  const int gN = (N + 255) / 256;
  const int gE = (E + 255) / 256;

  deg_init_kernel<<<gN, 256, 0, stream>>>(dinv, N);
  deg_count_kernel<<<gE, 256, 0, stream>>>(dstp, dinv, E);
  deg_rsqrt_kernel<<<gN, 256, 0, stream>>>(dinv, N);

  const int tiles = (N + 15) / 16;
  const int gemmBlocks = (tiles + 3) / 4;       // 4 waves of 32 per block
  const int gNH = (N * HDIM + 255) / 256;
  const int gE16 = (int)(((size_t)E * 16 + 255) / 256);

  const float* hin[3]  = {x, h1, h2};
  const float* Wk[3]   = {W0, W1, W2};
  const float* bk[3]   = {b0, b1, b2};
  float*       hout[3] = {h1, h2, h3};

  for (int k = 0; k < 3; ++k) {
    gcn_gemm64_kernel<<<gemmBlocks, 128, 0, stream>>>(hin[k], Wk[k], tmp, N);
    gcn_init64_kernel<<<gNH, 256, 0, stream>>>(tmp, dinv, bk[k], hout[k], N);
    gcn_scatter64_kernel<<<gE16, 256, 0, stream>>>(srcp, dstp, dinv, tmp, hout[k], E);
  }

  layer4_kernel<<<gN, 256, 0, stream>>>(h3, W3, b3, dinv, t1, h4, N);
  scatter1_kernel<<<gE, 256, 0, stream>>>(srcp, dstp, dinv, t1, h4, E);

  const int gOUT = (int)(((size_t)N * 16 + 255) / 256);
  conv_kernel<<<gOUT, 256, 0, stream>>>(h1, h2, h3, h4, cw, cb, out, N);
}